// QLSTMTensorRing_10917806866660
// MI455X (gfx1250) — compile-verified
//
#include <hip/hip_runtime.h>
#include <hip/hip_bf16.h>
#include <math.h>

#define B_SZ 256
#define T_SZ 512
#define F_SZ 128
#define H_SZ 128
#define NQ   4
#define QDIM 16

typedef __attribute__((ext_vector_type(2))) float v2f;
typedef __attribute__((ext_vector_type(4))) float v4f;
typedef __attribute__((ext_vector_type(8))) float v8f;

// ---------------------------------------------------------------------------
// Kernel A: yx[b*T+t][q] = Win_b[q] + sum_f x[b,t,f] * Win_w[q, H_SZ+f]
// One wave per 16-row tile; V_WMMA_F32_16X16X4_F32 over K=128 (32 steps).
// Memory bound (64MB of x @ 23.3TB/s ~ 2.8us floor). B-operand padding is
// done with an unconditional clamped load * 0/1 mask so EXEC stays all-1s
// and no divergent-load branches appear in the K loop.
// ---------------------------------------------------------------------------
__global__ __launch_bounds__(256) void qlstm_precompute_yx(
    const float* __restrict__ x, const float* __restrict__ Win_w,
    const float* __restrict__ Win_b, float* __restrict__ yx)
{
    const int lane = threadIdx.x & 31;
    const int wave = threadIdx.x >> 5;
    const int tile = blockIdx.x * 8 + wave;          // 16 (b,t)-rows per tile
    const long row0 = (long)tile * 16;
    const int  m    = lane & 15;                     // A: row, B/D: column
    const int  koff = (lane >> 4) * 2;               // f32 A/B lane-half K split

    const float* xrow = x + (row0 + m) * F_SZ + koff;
    // Clamped row => valid address for every lane; zero pad via mask multiply.
    const float* wrow = Win_w + (m & 3) * (F_SZ + H_SZ) + H_SZ + koff;
    const float  msk  = (m < NQ) ? 1.0f : 0.0f;

    v8f acc = {};
    #pragma unroll 4
    for (int k = 0; k < F_SZ; k += 4) {
        v2f a = __builtin_nontemporal_load((const v2f*)(xrow + k)); // read-once
        v2f b;
        b.x = wrow[k]     * msk;
        b.y = wrow[k + 1] * msk;
        acc = __builtin_amdgcn_wmma_f32_16x16x4_f32(
                  false, a, false, b, (short)0, acc, false, false);
    }
    if (m < NQ) {
        const float bias  = Win_b[m];
        const int   mbase = (lane >> 4) * 8;         // C/D: lanes>=16 hold M=8..15
        #pragma unroll
        for (int r = 0; r < 8; ++r)
            yx[(row0 + mbase + r) * NQ + m] = acc[r] + bias;   // RT: reused by scan
    }
}

// ---------------------------------------------------------------------------
// 4-qubit statevector circuit, fully register-resident (16 complex amps).
// qubit q <-> bit (3-q) of the state index (qubit 0 = MSB, per reference).
// ---------------------------------------------------------------------------
__device__ __forceinline__ void vqc4(const float ry[NQ], const float rz[NQ],
                                     const float wg[NQ], float e[NQ])
{
    float ar[QDIM], ai[QDIM];
    #pragma unroll
    for (int i = 0; i < QDIM; ++i) { ar[i] = 0.0f; ai[i] = 0.0f; }
    ar[0] = 1.0f;
    const float S2 = 0.70710678118654752440f;

    // Per qubit: H, RY(ry), RZ(rz) fused on each (i0,i1) pair
    #pragma unroll
    for (int q = 0; q < NQ; ++q) {
        const int bit = 1 << (3 - q);
        float sy, cy; __sincosf(ry[q] * 0.5f, &sy, &cy);
        float sz, cz; __sincosf(rz[q] * 0.5f, &sz, &cz);
        #pragma unroll
        for (int i = 0; i < QDIM; ++i) {
            if (i & bit) continue;
            const int j = i | bit;
            // H
            float h0r = S2 * (ar[i] + ar[j]), h0i = S2 * (ai[i] + ai[j]);
            float h1r = S2 * (ar[i] - ar[j]), h1i = S2 * (ai[i] - ai[j]);
            // RY
            float r0r = cy * h0r - sy * h1r, r0i = cy * h0i - sy * h1i;
            float r1r = sy * h0r + cy * h1r, r1i = sy * h0i + cy * h1i;
            // RZ: a0 *= (cz - i sz), a1 *= (cz + i sz)
            ar[i] = r0r * cz + r0i * sz;  ai[i] = r0i * cz - r0r * sz;
            ar[j] = r1r * cz - r1i * sz;  ai[j] = r1i * cz + r1r * sz;
        }
    }
    // Ring RY layer
    #pragma unroll
    for (int q = 0; q < NQ; ++q) {
        const int bit = 1 << (3 - q);
        float sw, cw; __sincosf(wg[q] * 0.5f, &sw, &cw);
        #pragma unroll
        for (int i = 0; i < QDIM; ++i) {
            if (i & bit) continue;
            const int j = i | bit;
            float a0r = ar[i], a0i = ai[i], a1r = ar[j], a1i = ai[j];
            ar[i] = cw * a0r - sw * a1r;  ai[i] = cw * a0i - sw * a1i;
            ar[j] = sw * a0r + cw * a1r;  ai[j] = sw * a0i + cw * a1i;
        }
    }
    // Ring CNOTs: control q, target (q+1)%4
    #pragma unroll
    for (int q = 0; q < NQ; ++q) {
        const int cb = 1 << (3 - q);
        const int tb = 1 << (3 - ((q + 1) & 3));
        #pragma unroll
        for (int i = 0; i < QDIM; ++i) {
            if ((i & cb) && !(i & tb)) {
                const int j = i | tb;
                float tr = ar[i]; ar[i] = ar[j]; ar[j] = tr;
                float ti = ai[i]; ai[i] = ai[j]; ai[j] = ti;
            }
        }
    }
    // PauliZ expectation values
    #pragma unroll
    for (int q = 0; q < NQ; ++q) e[q] = 0.0f;
    #pragma unroll
    for (int i = 0; i < QDIM; ++i) {
        const float p = ar[i] * ar[i] + ai[i] * ai[i];
        #pragma unroll
        for (int q = 0; q < NQ; ++q)
            e[q] += ((i >> (3 - q)) & 1) ? -p : p;
    }
}

__device__ __forceinline__ float sigmoidf_(float x)
{
    return 1.0f / (1.0f + __expf(-x));
}

// ---------------------------------------------------------------------------
// Kernel B: the whole T=512 scan. One wave32 per batch row (rows are fully
// independent). All state in VGPRs; h.Win_h reduced via shfl_xor butterfly;
// each lane redundantly simulates circuit for gate (lane&3), broadcast via
// shfl; coalesced non-temporal b128 stores of h each step (write-once 64MB).
// ---------------------------------------------------------------------------
__global__ __launch_bounds__(256) void qlstm_scan(
    const float* __restrict__ Win_w,  const float* __restrict__ Wout_w,
    const float* __restrict__ Wout_b,
    const float* __restrict__ wf, const float* __restrict__ wi,
    const float* __restrict__ wu, const float* __restrict__ wo,
    const float* __restrict__ yx, float* __restrict__ out)
{
    const int lane = threadIdx.x & 31;
    const int b    = blockIdx.x * 8 + (threadIdx.x >> 5);
    if (b >= B_SZ) return;
    const int g = lane & 3;                      // gate simulated by this lane

    // Preload per-lane weight slices (all loop-invariant)
    float wq[NQ][4];                              // Win_h[q, lane*4+j]
    #pragma unroll
    for (int q = 0; q < NQ; ++q)
        #pragma unroll
        for (int j = 0; j < 4; ++j)
            wq[q][j] = Win_w[q * (F_SZ + H_SZ) + lane * 4 + j];

    float wout[4][NQ], woutb[4];
    #pragma unroll
    for (int j = 0; j < 4; ++j) {
        woutb[j] = Wout_b[lane * 4 + j];
        #pragma unroll
        for (int q = 0; q < NQ; ++q)
            wout[j][q] = Wout_w[(lane * 4 + j) * NQ + q];
    }
    const float* wsel = (g == 0) ? wf : (g == 1) ? wi : (g == 2) ? wu : wo;
    float wg[NQ];
    #pragma unroll
    for (int q = 0; q < NQ; ++q) wg[q] = wsel[q];

    float h[4] = {0.f, 0.f, 0.f, 0.f};
    float c[4] = {0.f, 0.f, 0.f, 0.f};

    float* hseq = out;
    float* hout = out + (size_t)B_SZ * T_SZ * H_SZ;
    float* cout = hout + (size_t)B_SZ * H_SZ;
    const float* yxb = yx + (size_t)b * T_SZ * NQ;

    for (int t = 0; t < T_SZ; ++t) {
        // y[q] = h . Win_h[q,:] + yx  (butterfly reduce over wave)
        float y[NQ];
        #pragma unroll
        for (int q = 0; q < NQ; ++q) {
            float p = h[0]*wq[q][0] + h[1]*wq[q][1] + h[2]*wq[q][2] + h[3]*wq[q][3];
            #pragma unroll
            for (int m = 16; m >= 1; m >>= 1) p += __shfl_xor(p, m, 32);
            y[q] = p + yxb[t * NQ + q];
        }
        float ry[NQ], rz[NQ];
        #pragma unroll
        for (int q = 0; q < NQ; ++q) { ry[q] = atanf(y[q]); rz[q] = atanf(y[q]*y[q]); }

        float e[NQ];
        vqc4(ry, rz, wg, e);

        // Broadcast per-gate expvals from lanes 0..3
        float eF[NQ], eI[NQ], eU[NQ], eO[NQ];
        #pragma unroll
        for (int q = 0; q < NQ; ++q) {
            eF[q] = __shfl(e[q], 0, 32);
            eI[q] = __shfl(e[q], 1, 32);
            eU[q] = __shfl(e[q], 2, 32);
            eO[q] = __shfl(e[q], 3, 32);
        }

        v4f hv;
        #pragma unroll
        for (int j = 0; j < 4; ++j) {
            float af = woutb[j], ag = woutb[j], au = woutb[j], ao = woutb[j];
            #pragma unroll
            for (int q = 0; q < NQ; ++q) {
                af += eF[q] * wout[j][q];
                ag += eI[q] * wout[j][q];
                au += eU[q] * wout[j][q];
                ao += eO[q] * wout[j][q];
            }
            const float fg = sigmoidf_(af), ig = sigmoidf_(ag);
            const float gg = tanhf(au),     og = sigmoidf_(ao);
            c[j] = fg * c[j] + ig * gg;
            h[j] = og * tanhf(c[j]);
            hv[j] = h[j];
        }
        // Write-once 64MB stream: non-temporal b128 store
        __builtin_nontemporal_store(hv,
            (v4f*)(hseq + ((size_t)b * T_SZ + t) * H_SZ + lane * 4));
    }

    #pragma unroll
    for (int j = 0; j < 4; ++j) {
        hout[(size_t)b * H_SZ + lane * 4 + j] = h[j];
        cout[(size_t)b * H_SZ + lane * 4 + j] = c[j];
    }
}

// ---------------------------------------------------------------------------
extern "C" void kernel_launch(void* const* d_in, const int* in_sizes, int n_in,
                              void* d_out, int out_size, void* d_ws, size_t ws_size,
                              hipStream_t stream)
{
    const float* x      = (const float*)d_in[0];
    const float* Win_w  = (const float*)d_in[1];
    const float* Win_b  = (const float*)d_in[2];
    const float* Wout_w = (const float*)d_in[3];
    const float* Wout_b = (const float*)d_in[4];
    const float* wf     = (const float*)d_in[5];
    const float* wi     = (const float*)d_in[6];
    const float* wu     = (const float*)d_in[7];
    const float* wo     = (const float*)d_in[8];

    float* yx = (float*)d_ws;                 // B*T*NQ floats = 2 MB scratch

    // A: B*T/16 = 8192 tiles, 8 waves/block -> 1024 blocks
    hipLaunchKernelGGL(qlstm_precompute_yx,
                       dim3((B_SZ * T_SZ / 16) / 8), dim3(256), 0, stream,
                       x, Win_w, Win_b, yx);
    // B: one wave per batch row, 8 waves/block -> 32 blocks
    hipLaunchKernelGGL(qlstm_scan,
                       dim3(B_SZ / 8), dim3(256), 0, stream,
                       Win_w, Wout_w, Wout_b, wf, wi, wu, wo, yx, (float*)d_out);
}